// EM_step_86466281603856
// MI455X (gfx1250) — compile-verified
//
#include <hip/hip_runtime.h>

// Fused EM step for MI455X (gfx1250, wave32, WMMA).
//   E-step: dist = (|f|^2 - 2 f.m + |m|^2)*valid + adj ; p = softmax_k(-dist)
//   M-step: out[k,d] = (sum_n p[n,k] f[n,d]) / max(sum_n p[n,k], 1e-12)
// Both contractions run on V_WMMA_F32_16X16X4_F32. Posteriors live in LDS only;
// valid/adjust (268 MB) streamed with NT hints; softmax state kept off VGPRs
// to stay under 256 registers for occupancy. 512 WGs (2 per WGP at ~155 KB LDS)
// to cover the full shader-engine array.

#define B_ 8
#define N_ 16384
#define K_ 256
#define D_ 20
#define NCHUNKS 64               // workgroups per batch -> 512 WGs total
#define NB 128                   // pixels per iteration per WG (8 waves x 16)
#define NBP 130                  // padded LDS row stride for p (floats)
#define ITERS ((N_ / NCHUNKS) / NB)   // 2

typedef __attribute__((ext_vector_type(2))) float v2f;
typedef __attribute__((ext_vector_type(8))) float v8f;

__global__ __launch_bounds__(256) void em_zero(float* __restrict__ out,
                                               float* __restrict__ den) {
  int i = blockIdx.x * 256 + threadIdx.x;
  if (i < B_ * K_ * D_) out[i] = 0.0f;
  if (i < B_ * K_) den[i] = 0.0f;
}

__global__ __launch_bounds__(256) void em_main(const float* __restrict__ mean,
                                               const float* __restrict__ feat,
                                               const float* __restrict__ valid,
                                               const float* __restrict__ adj,
                                               float* __restrict__ out,
                                               float* __restrict__ den) {
  __shared__ float smean[K_ * D_];   // mean tile for this batch, [k][d]
  __shared__ float smsq[K_];         // |mean_k|^2
  __shared__ float pbuf[K_ * NBP];   // -dist -> exp(.), transposed [k][n_local]
  __shared__ float rinvb[NB];        // 1/rowsum per block-local pixel

  const int b = blockIdx.x / NCHUNKS;
  const int chunk = blockIdx.x % NCHUNKS;
  const int tid = threadIdx.x;
  const int w = tid >> 5;           // wave id 0..7
  const int lane = tid & 31;
  const int hi = lane >> 4;         // lane half
  const int lr = lane & 15;

  // ---- stage mean into LDS, compute mean_sq (one k-row per thread) ----
  {
    const float* mrow = mean + ((size_t)b * K_ + tid) * D_;
    float sq = 0.0f;
#pragma unroll
    for (int d = 0; d < D_; ++d) {
      float v = mrow[d];
      smean[tid * D_ + d] = v;
      sq += v * v;
    }
    smsq[tid] = sq;
  }
  __syncthreads();

  // persistent M-step accumulators: wave owns k-tiles {2w, 2w+1}, d-tiles {0,1}
  v8f acc[2][2];
#pragma unroll
  for (int q = 0; q < 2; ++q)
#pragma unroll
    for (int dt = 0; dt < 2; ++dt) acc[q][dt] = (v8f){0,0,0,0,0,0,0,0};

  const int nchunk0 = chunk * (N_ / NCHUNKS);

  for (int it = 0; it < ITERS; ++it) {
    const int nb0 = nchunk0 + it * NB;  // 128-pixel block base
    const int n0 = nb0 + w * 16;        // this wave's 16-pixel strip

    __syncthreads();  // pbuf/rinvb from previous iteration fully consumed

    // ---- A fragments: features strip, 16x4 f32 layout (V0:K0/K2, V1:K1/K3)
    v2f a[5];
    float psq = 0.0f;
    {
      const float* fbase = feat + ((size_t)b * N_ + n0 + lr) * D_ + 2 * hi;
#pragma unroll
      for (int s = 0; s < 5; ++s) {
        v2f t;
        t.x = fbase[s * 4];
        t.y = fbase[s * 4 + 1];
        a[s] = t;
        psq += t.x * t.x + t.y * t.y;
      }
    }
    // |f|^2 per strip row: each half holds 10 of the 20 d's
    float fsq = psq + __shfl_xor(psq, 16, 32);
    float fsqrow[8];
#pragma unroll
    for (int j = 0; j < 8; ++j) fsqrow[j] = __shfl(fsq, j + 8 * hi, 32);

    // ---- E-step pass 1: -dist into pbuf (transposed), track row max ----
    float rmax[8];
#pragma unroll
    for (int j = 0; j < 8; ++j) rmax[j] = -3.0e38f;

#pragma unroll
    for (int kt = 0; kt < 16; ++kt) {
      v2f bm[5];  // B fragments: mean^T, rows d(4) x cols k(16)
      {
        const float* mrow = smean + (kt * 16 + lr) * D_ + 2 * hi;
#pragma unroll
        for (int s = 0; s < 5; ++s) {
          v2f t;
          t.x = mrow[s * 4];
          t.y = mrow[s * 4 + 1];
          bm[s] = t;
        }
      }
      v8f c = (v8f){0,0,0,0,0,0,0,0};
#pragma unroll
      for (int s = 0; s < 5; ++s)
        c = __builtin_amdgcn_wmma_f32_16x16x4_f32(false, a[s], false, bm[s],
                                                  (short)0, c, false, false);
      const float msq = smsq[kt * 16 + lr];
      const size_t vbase = ((size_t)b * N_ + n0) * K_ + kt * 16 + lr;
      const int prow = (kt * 16 + lr) * NBP + w * 16;
#pragma unroll
      for (int j = 0; j < 8; ++j) {
        const int m = j + 8 * hi;  // strip row for this C element
        const float vs = __builtin_nontemporal_load(valid + vbase + (size_t)m * K_);
        const float ad = __builtin_nontemporal_load(adj + vbase + (size_t)m * K_);
        const float dist = (fsqrow[j] - 2.0f * c[j] + msq) * vs + ad;
        const float sc = -dist;    // 2*STD^2 == 1
        rmax[j] = fmaxf(rmax[j], sc);
        pbuf[prow + m] = sc;
      }
    }

    // ---- row max across lanes (within each 16-lane half) ----
#pragma unroll
    for (int j = 0; j < 8; ++j) {
      float r = rmax[j];
      r = fmaxf(r, __shfl_xor(r, 1, 32));
      r = fmaxf(r, __shfl_xor(r, 2, 32));
      r = fmaxf(r, __shfl_xor(r, 4, 32));
      r = fmaxf(r, __shfl_xor(r, 8, 32));
      rmax[j] = r;
    }

    // ---- E-step pass 2: exp in place, accumulate row sums ----
    float rsum[8];
#pragma unroll
    for (int j = 0; j < 8; ++j) rsum[j] = 0.0f;
#pragma unroll
    for (int kt = 0; kt < 16; ++kt) {
      const int prow = (kt * 16 + lr) * NBP + w * 16;
#pragma unroll
      for (int j = 0; j < 8; ++j) {
        const int m = j + 8 * hi;
        const float e = __expf(pbuf[prow + m] - rmax[j]);
        rsum[j] += e;
        pbuf[prow + m] = e;
      }
    }
#pragma unroll
    for (int j = 0; j < 8; ++j) {
      float r = rsum[j];
      r += __shfl_xor(r, 1, 32);
      r += __shfl_xor(r, 2, 32);
      r += __shfl_xor(r, 4, 32);
      r += __shfl_xor(r, 8, 32);
      // r >= 1 (max term is exp(0)); publish 1/rowsum for this strip row
      if (lr == j) rinvb[w * 16 + j + 8 * hi] = 1.0f / r;
    }
    __syncthreads();

    // ---- M-step GEMM: acc += E^T (k x n) * [rinv*F | rinv] (n x d) ----
#pragma unroll 2
    for (int ns = 0; ns < NB / 4; ++ns) {
      v2f pa[2];  // A frags: rows k(16), cols n(4), unnormalized exp
#pragma unroll
      for (int q = 0; q < 2; ++q) {
        const int kk = (w * 2 + q) * 16 + lr;
        const float* pr = pbuf + kk * NBP + ns * 4 + 2 * hi;
        v2f t;
        t.x = pr[0];
        t.y = pr[1];
        pa[q] = t;
      }
      // fold 1/rowsum into the feature fragments (rows are pixels)
      const float ri0 = rinvb[ns * 4 + 2 * hi];
      const float ri1 = rinvb[ns * 4 + 2 * hi + 1];
      v2f fb[2];  // B frags: rows n(4), cols d(16); col 20 = denominator
#pragma unroll
      for (int dt = 0; dt < 2; ++dt) {
        const int d = dt * 16 + lr;
        float x0, x1;
        if (d < D_) {
          const float* f0 =
              feat + ((size_t)b * N_ + nb0 + ns * 4 + 2 * hi) * D_ + d;
          x0 = f0[0] * ri0;
          x1 = f0[D_] * ri1;
        } else if (d == D_) {
          x0 = ri0;
          x1 = ri1;
        } else {
          x0 = 0.0f;
          x1 = 0.0f;
        }
        v2f t;
        t.x = x0;
        t.y = x1;
        fb[dt] = t;
      }
#pragma unroll
      for (int q = 0; q < 2; ++q)
#pragma unroll
        for (int dt = 0; dt < 2; ++dt)
          acc[q][dt] = __builtin_amdgcn_wmma_f32_16x16x4_f32(
              false, pa[q], false, fb[dt], (short)0, acc[q][dt], false, false);
    }
  }

  // ---- flush partial numerator/denominator (L2-resident atomics) ----
#pragma unroll
  for (int q = 0; q < 2; ++q) {
    const int kbase = (w * 2 + q) * 16;
#pragma unroll
    for (int dt = 0; dt < 2; ++dt) {
      const int d = dt * 16 + lr;
#pragma unroll
      for (int j = 0; j < 8; ++j) {
        const int k = kbase + j + 8 * hi;
        const float v = acc[q][dt][j];
        if (d < D_)
          atomicAdd(&out[((size_t)b * K_ + k) * D_ + d], v);
        else if (d == D_)
          atomicAdd(&den[b * K_ + k], v);
      }
    }
  }
}

__global__ __launch_bounds__(256) void em_fin(float* __restrict__ out,
                                              const float* __restrict__ den) {
  int i = blockIdx.x * 256 + threadIdx.x;
  if (i < B_ * K_ * D_) out[i] = out[i] / fmaxf(den[i / D_], 1e-12f);
}

extern "C" void kernel_launch(void* const* d_in, const int* in_sizes, int n_in,
                              void* d_out, int out_size, void* d_ws,
                              size_t ws_size, hipStream_t stream) {
  (void)in_sizes; (void)n_in; (void)out_size; (void)ws_size;
  const float* mean = (const float*)d_in[0];
  const float* feat = (const float*)d_in[1];
  const float* valid = (const float*)d_in[2];
  const float* adj = (const float*)d_in[3];
  float* out = (float*)d_out;
  float* den = (float*)d_ws;  // 2048 floats of scratch for denominators

  const int tot = B_ * K_ * D_;
  em_zero<<<(tot + 255) / 256, 256, 0, stream>>>(out, den);
  em_main<<<B_ * NCHUNKS, 256, 0, stream>>>(mean, feat, valid, adj, out, den);
  em_fin<<<(tot + 255) / 256, 256, 0, stream>>>(out, den);
}